// GCNConv_Decoder_22316650070982
// MI455X (gfx1250) — compile-verified
//
#include <hip/hip_runtime.h>

// ---------------------------------------------------------------------------
// 2-layer GCN for MI455X (gfx1250, wave32).
//   h  = X @ W                         (f32 WMMA 16x16x4, full precision)
//   agg[dst] += h[src] * dinv[s]*dinv[d]   (L2-resident gather + f32 atomics)
//   out = agg + h * dinv^2 + b
// Entire working set (~52MB) fits in the 192MB L2, so edge traffic never
// leaves the cache; design keeps lane-contiguous feature access per edge.
// ---------------------------------------------------------------------------

typedef __attribute__((ext_vector_type(2))) float v2f;
typedef __attribute__((ext_vector_type(8))) float v8f;

#define GCN_NODES 50000
#define GCN_EDGES 800000
#define F_IN  96
#define F_MID 96
#define F_OUT 64

// ---------------- utility: zero fill ----------------
__global__ void zero_f32(float* __restrict__ p, int n) {
    int i = blockIdx.x * blockDim.x + threadIdx.x;
    if (i < n) p[i] = 0.0f;
}

// ---------------- degree count (edges at dst) ----------------
__global__ void degree_kernel(const long long* __restrict__ dst,
                              float* __restrict__ deg, int nE) {
    int i = blockIdx.x * blockDim.x + threadIdx.x;
    if (i < nE) atomicAdd(&deg[(int)dst[i]], 1.0f);
}

// ---------------- dinv = rsqrt(deg + 1) ----------------
__global__ void dinv_kernel(float* __restrict__ deg, int n) {
    int i = blockIdx.x * blockDim.x + threadIdx.x;
    if (i < n) deg[i] = rsqrtf(deg[i] + 1.0f);
}

// ---------------- f32 WMMA GEMM: C[M x NCOL] = A[M x K] @ B[K x NCOL] -------
// One wave32 per 16x16 output tile. M must be a multiple of 16 (50000 = 3125*16).
// A-frag (16x4, MxK): lane%16 = M row; half-wave (lane/16) picks K pair; 2 VGPRs.
// B-frag (4x16, KxN): symmetric — lane%16 = N col; half-wave picks K pair.
// C/D (16x16 f32): lane%16 = N col; row M = vgpr + 8*(lane/16).
template <int K, int NCOL>
__global__ void gemm_wmma_f32(const float* __restrict__ A,
                              const float* __restrict__ B,
                              float* __restrict__ C, int M) {
    const int wavesPerBlock = blockDim.x >> 5;
    const int wave = blockIdx.x * wavesPerBlock + (threadIdx.x >> 5);
    const int lane = threadIdx.x & 31;
    constexpr int colTiles = NCOL / 16;
    const int rowTile = wave / colTiles;
    const int colTile = wave % colTiles;
    if (rowTile * 16 >= M) return;           // wave-uniform: EXEC stays all-1s

    const int half = lane >> 4;              // 0 or 1 (selects K pair)
    const int lid  = lane & 15;
    const int row  = rowTile * 16 + lid;     // A row for this lane
    const int col  = colTile * 16 + lid;     // B/C column for this lane

    const float* __restrict__ Arow = A + (long)row * K;

    v8f c = {};
#pragma unroll
    for (int k = 0; k < K; k += 4) {
        const int ka = k + 2 * half;
        v2f a, b;
        a.x = Arow[ka];
        a.y = Arow[ka + 1];
        b.x = B[(long)ka * NCOL + col];
        b.y = B[(long)(ka + 1) * NCOL + col];
        // D = A*B + C, full fp32 (V_WMMA_F32_16X16X4_F32)
        c = __builtin_amdgcn_wmma_f32_16x16x4_f32(
                /*neg_a=*/false, a, /*neg_b=*/false, b,
                /*c_mod=*/(short)0, c, /*reuse_a=*/false, /*reuse_b=*/false);
    }

#pragma unroll
    for (int v = 0; v < 8; ++v) {
        const int m = rowTile * 16 + v + 8 * half;
        C[(long)m * NCOL + col] = c[v];
    }
}

// ---------------- per-edge gather + normalized scatter-add ----------------
// One wave32 per edge; F/32 floats per lane, lane-contiguous (coalesced,
// 128B per wave per chunk). h and agg are L2-resident (19MB << 192MB L2).
template <int F>
__global__ void scatter_edges(const long long* __restrict__ src,
                              const long long* __restrict__ dst,
                              const float* __restrict__ dinv,
                              const float* __restrict__ h,
                              float* __restrict__ agg, int nE) {
    const int wave = blockIdx.x * (blockDim.x >> 5) + (threadIdx.x >> 5);
    const int lane = threadIdx.x & 31;
    if (wave >= nE) return;
    const int s = (int)src[wave];
    const int d = (int)dst[wave];
    const float norm = dinv[s] * dinv[d];
    const float* __restrict__ hs = h + (long)s * F;
    float* __restrict__ ad = agg + (long)d * F;
#pragma unroll
    for (int c = 0; c < F / 32; ++c) {
        const int j = lane + 32 * c;
        atomicAdd(&ad[j], hs[j] * norm);
    }
}

// ---------------- finalize: agg += h * dinv^2 + bias ----------------
template <int F>
__global__ void finalize_layer(const float* __restrict__ h,
                               const float* __restrict__ dinv,
                               const float* __restrict__ bias,
                               float* __restrict__ agg, int n) {
    const long i = (long)blockIdx.x * blockDim.x + threadIdx.x;
    const long total = (long)n * F;
    if (i >= total) return;
    const int node = (int)(i / F);
    const int j = (int)(i % F);
    const float di = dinv[node];
    agg[i] += h[i] * di * di + bias[j];
}

// ---------------------------------------------------------------------------
extern "C" void kernel_launch(void* const* d_in, const int* in_sizes, int n_in,
                              void* d_out, int out_size, void* d_ws, size_t ws_size,
                              hipStream_t stream) {
    const float*     x   = (const float*)d_in[0];
    const long long* ei  = (const long long*)d_in[1];   // int64 [2, E]
    const float*     W1  = (const float*)d_in[2];
    const float*     b1  = (const float*)d_in[3];
    const float*     W2  = (const float*)d_in[4];
    const float*     b2  = (const float*)d_in[5];
    float*           out = (float*)d_out;

    const long long* src = ei;
    const long long* dst = ei + GCN_EDGES;

    // workspace layout (floats), 128B-aligned sections
    float* ws   = (float*)d_ws;
    float* deg  = ws;                                  // 50000 (becomes dinv)
    float* h1   = ws + 50048;                          // 50000*96
    float* agg1 = h1 + (long)GCN_NODES * F_MID;        // 50000*96
    float* h2   = agg1 + (long)GCN_NODES * F_MID;      // 50000*64

    const int nH1  = GCN_NODES * F_MID;                // 4,800,000
    const int nOut = GCN_NODES * F_OUT;                // 3,200,000

    // 1) zero accumulators (atomics add into them every call)
    zero_f32<<<(GCN_NODES + 255) / 256, 256, 0, stream>>>(deg, GCN_NODES);
    zero_f32<<<(nH1 + 255) / 256, 256, 0, stream>>>(agg1, nH1);
    zero_f32<<<(nOut + 255) / 256, 256, 0, stream>>>(out, nOut);

    // 2) degree + dinv
    degree_kernel<<<(GCN_EDGES + 255) / 256, 256, 0, stream>>>(dst, deg, GCN_EDGES);
    dinv_kernel<<<(GCN_NODES + 255) / 256, 256, 0, stream>>>(deg, GCN_NODES);

    // 3) h1 = x @ W1  (3125 row-tiles * 6 col-tiles = 18750 waves, 8 waves/block)
    {
        const int tiles = (GCN_NODES / 16) * (F_MID / 16);
        const int blocks = (tiles + 7) / 8;
        gemm_wmma_f32<F_IN, F_MID><<<blocks, 256, 0, stream>>>(x, W1, h1, GCN_NODES);
    }

    // 4) edge aggregation layer 1 (one wave per edge)
    scatter_edges<F_MID><<<(GCN_EDGES + 7) / 8, 256, 0, stream>>>(
        src, dst, deg, h1, agg1, GCN_EDGES);

    // 5) agg1 += h1 * dinv^2 + b1
    finalize_layer<F_MID><<<(nH1 + 255) / 256, 256, 0, stream>>>(
        h1, deg, b1, agg1, GCN_NODES);

    // 6) h2 = agg1 @ W2  (3125 * 4 = 12500 waves)
    {
        const int tiles = (GCN_NODES / 16) * (F_OUT / 16);
        const int blocks = (tiles + 7) / 8;
        gemm_wmma_f32<F_MID, F_OUT><<<blocks, 256, 0, stream>>>(agg1, W2, h2, GCN_NODES);
    }

    // 7) edge aggregation layer 2 -> out
    scatter_edges<F_OUT><<<(GCN_EDGES + 7) / 8, 256, 0, stream>>>(
        src, dst, deg, h2, out, GCN_EDGES);

    // 8) out += h2 * dinv^2 + b2
    finalize_layer<F_OUT><<<(nOut + 255) / 256, 256, 0, stream>>>(
        h2, deg, b2, out, GCN_NODES);
}